// QueryAndGroup_81844896792773
// MI455X (gfx1250) — compile-verified
//
#include <hip/hip_runtime.h>

// Problem constants (match reference)
#define B_   4
#define N_   16384
#define M_   4096
#define C_   64
#define S_   32          // NSAMPLE
#define R2_  (0.4f * 0.4f)

typedef __attribute__((ext_vector_type(2))) float v2f;
typedef __attribute__((ext_vector_type(8))) float v8f;

// ---------------------------------------------------------------------------
// Kernel A: ball query via V_WMMA_F32_16X16X4_F32 + ballot-based selection.
// One wave (32 lanes) per 16 queries. d2 = qq + pp - 2*q.p as a K=4 GEMM:
//   A[m] = (x, y, z, 1)            (16x4)
//   B[:,n] = (-2x, -2y, -2z, pp)   (4x16)
//   C[m,n] = qq[m]
// Operand layouts (ISA 7.12.2, f32):
//   A 16x4 : VGPR0 = K0 (lanes0-15) / K2 (lanes16-31); VGPR1 = K1 / K3
//   B 4x16 : VGPR0 = K0 row (lanes0-15) / K1 row (lanes16-31); VGPR1 = K2 / K3
//   C/D    : VGPR r, lane l -> M = r + 8*(l>=16), N = l&15
//
// Point loads are software-pipelined 4 deep (64 points per group) so one
// memory-latency window is amortized over four WMMA+select bodies
// (only ~1 wave/SIMD exists to hide latency otherwise).
//
// Selection: ballot(d[r] < R2) is wave-uniform; bits[15:0] = row r hit mask,
// bits[31:16] = row r+8 hit mask. Lane m extracts its row mask with a
// branchless cndmask/or chain and appends hits with a ctz loop.
// ---------------------------------------------------------------------------
__global__ __launch_bounds__(32) void ball_query_wmma(
    const float* __restrict__ xyz,      // (B, N, 3)
    const float* __restrict__ new_xyz,  // (B, M, 3)
    int*         __restrict__ idx_out)  // (B, M, S)
{
  __shared__ float qb[16 * 4];       // per-query x,y,z,qq (SoA: qb[k*16+m])
  __shared__ int   idxbuf[16 * 33];  // per-row index lists, stride 33 (banking)

  const int lane = threadIdx.x;              // 0..31
  const int t    = blockIdx.x;               // query-tile id
  const int b    = t / (M_ / 16);
  const int m0   = (t % (M_ / 16)) * 16;

  // Stage the 16 queries into LDS.
  if (lane < 16) {
    const float* q = new_xyz + ((size_t)b * M_ + m0 + lane) * 3;
    float x = q[0], y = q[1], z = q[2];
    qb[0 * 16 + lane] = x;
    qb[1 * 16 + lane] = y;
    qb[2 * 16 + lane] = z;
    qb[3 * 16 + lane] = x * x + y * y + z * z;
  }
  __syncthreads();

  const int hi = lane >> 4;   // lane-half
  const int lo = lane & 15;
  const int k  = lane & 7;    // loop-invariant select index (hoistable cmps)

  // A operand (loop-invariant).
  v2f amat;
  amat.x = hi ? qb[2 * 16 + lo] : qb[0 * 16 + lo];  // K2=z : K0=x
  amat.y = hi ? 1.0f            : qb[1 * 16 + lo];  // K3=1 : K1=y

  // C operand: qq[row], row = r + 8*hi.
  v8f cmat;
#pragma unroll
  for (int r = 0; r < 8; ++r) cmat[r] = qb[3 * 16 + (r + 8 * hi)];

  // Lane m (<16) owns query row m; lanes 16-31 are "done" from the start.
  int mycnt = (lane < 16) ? 0 : S_;

  const float* xb = xyz + (size_t)b * N_ * 3;

  for (int n0 = 0; n0 < N_; n0 += 64) {
    // ---- Stage 1: issue 4 independent 16-point tile loads (pipelined). ----
    float px[4], py[4], pz[4];
#pragma unroll
    for (int s = 0; s < 4; ++s) {
      const float* p = xb + (size_t)(n0 + 16 * s + lo) * 3;
      px[s] = p[0]; py[s] = p[1]; pz[s] = p[2];
    }
    // Prefetch the next 64-point group (global_prefetch_b8).
    __builtin_prefetch(xb + (size_t)(n0 + 64 + lo) * 3, 0, 3);

    // ---- Stage 2: drain the pipeline, one WMMA + select per sub-tile. ----
#pragma unroll
    for (int s = 0; s < 4; ++s) {
      float pp = px[s] * px[s] + py[s] * py[s] + pz[s] * pz[s];
      v2f bmat;
      bmat.x = hi ? -2.0f * py[s] : -2.0f * px[s];  // K1 : K0
      bmat.y = hi ? pp            : -2.0f * pz[s];  // K3 : K2

      // 16x16 d2 tile in one matrix op.
      v8f d = __builtin_amdgcn_wmma_f32_16x16x4_f32(
          false, amat, false, bmat, (short)0, cmat, false, false);

      // Hit masks: 8 uniform ballots cover all 16 rows (two rows each).
      unsigned bl[8];
#pragma unroll
      for (int r = 0; r < 8; ++r)
        bl[r] = (unsigned)__builtin_amdgcn_ballot_w32(d[r] < R2_);

      // Branchless per-lane row-mask extraction: OR-accumulate of
      // (k==r ? bl[r] : 0) -> v_cndmask + v_or chain, cmps hoisted.
      unsigned a = 0u;
#pragma unroll
      for (int r = 0; r < 8; ++r) a |= (k == r) ? bl[r] : 0u;
      unsigned mask = (lane < 8) ? (a & 0xFFFFu) : (a >> 16);
      if (lane >= 16) mask = 0u;

      // Append hits in ascending column order; iterates only over real hits.
      const int nbase = n0 + 16 * s;
      while (mask && mycnt < S_) {
        int j = __builtin_ctz(mask);
        mask &= mask - 1u;
        idxbuf[lane * 33 + mycnt] = nbase + j;
        ++mycnt;
      }
    }

    // Uniform early exit once every row has S_ hits (64-point granularity).
    if ((unsigned)__builtin_amdgcn_ballot_w32(mycnt >= S_) == 0xFFFFFFFFu)
      break;
  }

  // Pad: repeat first hit (or 0 if no hits) — matches reference semantics.
  if (lane < 16) {
    int first = (mycnt > 0) ? idxbuf[lane * 33] : 0;
    for (int kk = mycnt; kk < S_; ++kk) idxbuf[lane * 33 + kk] = first;
  }
  __syncthreads();

  // Coalesced 128B writes of the index lists.
#pragma unroll
  for (int r = 0; r < 16; ++r)
    idx_out[((size_t)b * M_ + m0 + r) * S_ + lane] = idxbuf[r * 33 + lane];
}

// ---------------------------------------------------------------------------
// Kernel B: gather + center + concat. One wave per query, lane = sample.
// Output (B, 3+C, M, S): lane-contiguous 128B stores per channel, issued
// non-temporally so the 140MB stream does not evict the L2-resident
// 16MB feature table.
// ---------------------------------------------------------------------------
__global__ __launch_bounds__(256) void group_gather(
    const float* __restrict__ xyz,      // (B, N, 3)
    const float* __restrict__ new_xyz,  // (B, M, 3)
    const float* __restrict__ feat,     // (B, C, N)
    const int*   __restrict__ idx,      // (B, M, S)
    float*       __restrict__ out)      // (B, 3+C, M, S)
{
  const int lane = threadIdx.x & 31;
  const int q    = blockIdx.x * 8 + (threadIdx.x >> 5);  // 0..B*M-1
  const int b    = q / M_;
  const int m    = q - b * M_;

  const int id = idx[(size_t)q * S_ + lane];

  const float* pb = xyz + (size_t)b * N_ * 3 + (size_t)id * 3;
  float px = pb[0], py = pb[1], pz = pb[2];
  const float* qc = new_xyz + (size_t)q * 3;   // same addr across wave: broadcast
  float qx = qc[0], qy = qc[1], qz = qc[2];

  const size_t chstride = (size_t)M_ * S_;
  float* ob = out + (size_t)b * (3 + C_) * chstride + (size_t)m * S_ + lane;

  __builtin_nontemporal_store(px - qx, ob + 0 * chstride);
  __builtin_nontemporal_store(py - qy, ob + 1 * chstride);
  __builtin_nontemporal_store(pz - qz, ob + 2 * chstride);

  const float* fb = feat + (size_t)b * C_ * N_ + id;
#pragma unroll 8
  for (int c = 0; c < C_; ++c) {
    float v = fb[(size_t)c * N_];                 // gather, L2-resident table
    __builtin_nontemporal_store(v, ob + (size_t)(3 + c) * chstride);
  }
}

// ---------------------------------------------------------------------------
extern "C" void kernel_launch(void* const* d_in, const int* in_sizes, int n_in,
                              void* d_out, int out_size, void* d_ws, size_t ws_size,
                              hipStream_t stream) {
  const float* xyz     = (const float*)d_in[0];  // (B, N, 3)
  const float* new_xyz = (const float*)d_in[1];  // (B, M, 3)
  const float* feat    = (const float*)d_in[2];  // (B, C, N)
  float*       out     = (float*)d_out;          // (B, 3+C, M, S)

  int* d_idx = (int*)d_ws;  // B*M*S*4 = 2 MB scratch

  // Phase 1: WMMA ball query. One wave per 16 queries.
  ball_query_wmma<<<(B_ * M_) / 16, 32, 0, stream>>>(xyz, new_xyz, d_idx);

  // Phase 2: gather/center/concat. One wave per query, 8 waves per block.
  group_gather<<<(B_ * M_) / 8, 256, 0, stream>>>(xyz, new_xyz, feat, d_idx, out);

  (void)in_sizes; (void)n_in; (void)out_size; (void)ws_size;
}